// MS_QCR_Model_89215060672683
// MI455X (gfx1250) — compile-verified
//
#include <hip/hip_runtime.h>

typedef __attribute__((ext_vector_type(2))) float v2f;
typedef __attribute__((ext_vector_type(8))) float v8f;

#define NQ 5
#define DIM 32

// ---------------------------------------------------------------------------
// Kernel 1: fold CNOT permutations into each layer's 32x32 complex gate matrix
// and store B = M^T (row-major [k][n], n stride 1) for direct WMMA B-fragment
// loads.  M[j][k] = K[comp(j)][k],  K = kron(rot(w[l,0]),...,rot(w[l,4])).
// ---------------------------------------------------------------------------
__global__ void build_gates(const float* __restrict__ res_w,
                            float* __restrict__ Bre, float* __restrict__ Bim) {
  int idx = blockIdx.x * blockDim.x + threadIdx.x;
  if (idx >= 3 * DIM * DIM) return;
  int l = idx >> 10;        // layer
  int k = (idx >> 5) & 31;  // K index (column of M)
  int n = idx & 31;         // N index (row j of M)

  // composite CNOT permutation: perm0[perm1[...[perm4[n]]]]
  int r = n;
#pragma unroll
  for (int i = NQ - 1; i >= 0; --i) {
    int t = (i + l + 1) % NQ;
    if ((r >> (4 - i)) & 1) r ^= 1 << (4 - t);
  }

  float re = 1.f, im = 0.f;
#pragma unroll
  for (int q = 0; q < NQ; ++q) {
    const float* w = res_w + (l * NQ + q) * 3;
    float phi = w[0], th = w[1], om = w[2];
    float ch = cosf(0.5f * th), sh = sinf(0.5f * th);
    int rb = (r >> (4 - q)) & 1;
    int cb = (k >> (4 - q)) & 1;
    float er, ei;
    if (rb == 0 && cb == 0) { float a = -0.5f * (phi + om); er =  cosf(a) * ch; ei =  sinf(a) * ch; }
    else if (rb == 0)       { float a =  0.5f * (phi - om); er = -cosf(a) * sh; ei = -sinf(a) * sh; }
    else if (cb == 0)       { float a = -0.5f * (phi - om); er =  cosf(a) * sh; ei =  sinf(a) * sh; }
    else                    { float a =  0.5f * (phi + om); er =  cosf(a) * ch; ei =  sinf(a) * ch; }
    float tre = re * er - im * ei;
    im = re * ei + im * er;
    re = tre;
  }
  Bre[idx] = re;
  Bim[idx] = im;
}

// ---------------------------------------------------------------------------
// Kernel 2: fused frontend + 3 complex-GEMM quantum layers (f32 WMMA) + readout
// 128 threads = 4 wave32 per block; 1 sample per thread; wave owns 32 samples.
// ---------------------------------------------------------------------------
#define ROWSTRIDE 34  // 32 amps + 2 pad (bank-conflict-free fragment access)

__global__ __launch_bounds__(128) void msqcr_kernel(
    const float* __restrict__ x_short, const float* __restrict__ x_long,
    const float* __restrict__ conv_s_w, const float* __restrict__ conv_s_b,
    const float* __restrict__ lin_s_w,  const float* __restrict__ lin_s_b,
    const float* __restrict__ conv_l_w, const float* __restrict__ conv_l_b,
    const float* __restrict__ lin_l_w,  const float* __restrict__ lin_l_b,
    const float* __restrict__ readout_w, const float* __restrict__ readout_b,
    const float* __restrict__ Bre, const float* __restrict__ Bim,
    float* __restrict__ out) {
  __shared__ float s_re[128 * ROWSTRIDE];
  __shared__ float s_im[128 * ROWSTRIDE];

  const int tid = threadIdx.x;
  const int s = blockIdx.x * 128 + tid;

  // ---------------- classical frontend (registers) ----------------
  float xs[7];
  xs[0] = 0.f; xs[6] = 0.f;
#pragma unroll
  for (int i = 0; i < 5; ++i) xs[i + 1] = x_short[s * 5 + i];
  float xl[24];
  xl[0] = xl[1] = 0.f; xl[22] = xl[23] = 0.f;
#pragma unroll
  for (int i = 0; i < 20; ++i) xl[i + 2] = x_long[s * 20 + i];

  float fs0 = lin_s_b[0], fs1 = lin_s_b[1];
#pragma unroll
  for (int c = 0; c < 8; ++c) {
    float w0 = conv_s_w[c * 3], w1 = conv_s_w[c * 3 + 1], w2 = conv_s_w[c * 3 + 2];
    float bb = conv_s_b[c];
#pragma unroll
    for (int p = 0; p < 5; ++p) {
      float v = bb + w0 * xs[p] + w1 * xs[p + 1] + w2 * xs[p + 2];
      v = fmaxf(v, 0.f);
      fs0 += lin_s_w[c * 5 + p] * v;
      fs1 += lin_s_w[40 + c * 5 + p] * v;
    }
  }
  float fl0 = lin_l_b[0], fl1 = lin_l_b[1];
#pragma unroll
  for (int c = 0; c < 8; ++c) {
    float w0 = conv_l_w[c * 5],     w1 = conv_l_w[c * 5 + 1], w2 = conv_l_w[c * 5 + 2];
    float w3 = conv_l_w[c * 5 + 3], w4 = conv_l_w[c * 5 + 4];
    float bb = conv_l_b[c];
#pragma unroll
    for (int q = 0; q < 10; ++q) {
      int p0 = 2 * q, p1 = 2 * q + 1;
      float v0 = bb + w0 * xl[p0] + w1 * xl[p0 + 1] + w2 * xl[p0 + 2] + w3 * xl[p0 + 3] + w4 * xl[p0 + 4];
      float v1 = bb + w0 * xl[p1] + w1 * xl[p1 + 1] + w2 * xl[p1 + 2] + w3 * xl[p1 + 3] + w4 * xl[p1 + 4];
      v0 = fmaxf(v0, 0.f); v1 = fmaxf(v1, 0.f);
      float v = fmaxf(v0, v1);
      fl0 += lin_l_w[c * 10 + q] * v;
      fl1 += lin_l_w[80 + c * 10 + q] * v;
    }
  }

  float qin[5];
  qin[0] = fs0; qin[1] = fs1; qin[2] = fl0; qin[3] = fl1;
  qin[4] = 0.25f * (fs0 + fs1 + fl0 + fl1);
  float cq[5], sq[5];
#pragma unroll
  for (int q = 0; q < 5; ++q) {
    float a = 0.5f * 3.14159265358979323846f * qin[q];
    cq[q] = cosf(a);
    sq[q] = sinf(a);
  }
  // initial product state (purely real) -> LDS
  {
    int base = tid * ROWSTRIDE;
#pragma unroll
    for (int k = 0; k < 32; ++k) {
      float a = 1.f;
#pragma unroll
      for (int q = 0; q < 5; ++q) a *= ((k >> (4 - q)) & 1) ? sq[q] : cq[q];
      s_re[base + k] = a;
      s_im[base + k] = 0.f;
    }
  }
  __syncthreads();

  // ---------------- quantum layers: complex GEMM via f32 WMMA ----------------
  const int lane  = tid & 31;
  const int wave  = tid >> 5;
  const int mrow  = lane & 15;
  const int khalf = lane >> 4;  // which half of the K quartet this lane holds

#pragma unroll
  for (int layer = 0; layer < 3; ++layer) {
    const float* bre = Bre + layer * 1024;
    const float* bim = Bim + layer * 1024;

    // preload ALL A-fragments (entire 32x32 state of this wave) -> registers,
    // so in-place write-back below is hazard-free.
    v2f Are[2][8], Aim[2][8];
#pragma unroll
    for (int mt = 0; mt < 2; ++mt) {
      int base = (wave * 32 + mt * 16 + mrow) * ROWSTRIDE;
#pragma unroll
      for (int ks = 0; ks < 8; ++ks) {
        int k0 = ks * 4 + khalf * 2;
        v2f ar; ar.x = s_re[base + k0]; ar.y = s_re[base + k0 + 1];
        v2f ai; ai.x = s_im[base + k0]; ai.y = s_im[base + k0 + 1];
        Are[mt][ks] = ar;
        Aim[mt][ks] = ai;
      }
    }

    v8f Cre[2][2], Cim[2][2];
#pragma unroll
    for (int nt = 0; nt < 2; ++nt) {
      int n = nt * 16 + (lane & 15);
      Cre[nt][0] = {}; Cre[nt][1] = {};
      Cim[nt][0] = {}; Cim[nt][1] = {};
#pragma unroll
      for (int ks = 0; ks < 8; ++ks) {
        int k0 = ks * 4 + khalf * 2;
        v2f br; br.x = bre[k0 * 32 + n]; br.y = bre[(k0 + 1) * 32 + n];
        v2f bi; bi.x = bim[k0 * 32 + n]; bi.y = bim[(k0 + 1) * 32 + n];
        v2f bin = -bi;  // VALU negate (F32 WMMA NEG bits are C-negate only)
#pragma unroll
        for (int mt = 0; mt < 2; ++mt) {
          // C_re += Ar*Br - Ai*Bi ; C_im += Ar*Bi + Ai*Br
          Cre[nt][mt] = __builtin_amdgcn_wmma_f32_16x16x4_f32(
              false, Are[mt][ks], false, br,  (short)0, Cre[nt][mt], false, false);
          Cre[nt][mt] = __builtin_amdgcn_wmma_f32_16x16x4_f32(
              false, Aim[mt][ks], false, bin, (short)0, Cre[nt][mt], false, false);
          Cim[nt][mt] = __builtin_amdgcn_wmma_f32_16x16x4_f32(
              false, Are[mt][ks], false, bi,  (short)0, Cim[nt][mt], false, false);
          Cim[nt][mt] = __builtin_amdgcn_wmma_f32_16x16x4_f32(
              false, Aim[mt][ks], false, br,  (short)0, Cim[nt][mt], false, false);
        }
      }
    }

    // write new state back in place (wave owns its 32 rows exclusively)
#pragma unroll
    for (int nt = 0; nt < 2; ++nt) {
      int n = nt * 16 + (lane & 15);
#pragma unroll
      for (int mt = 0; mt < 2; ++mt) {
#pragma unroll
        for (int r = 0; r < 8; ++r) {
          int m = r + 8 * khalf;
          int base = (wave * 32 + mt * 16 + m) * ROWSTRIDE;
          s_re[base + n] = Cre[nt][mt][r];
          s_im[base + n] = Cim[nt][mt][r];
        }
      }
    }
    __syncthreads();
  }

  // ---------------- probs -> Z expectations -> readout ----------------
  float z0 = 0, z1 = 0, z2 = 0, z3 = 0, z4 = 0;
  {
    int base = tid * ROWSTRIDE;
#pragma unroll
    for (int k = 0; k < 32; ++k) {
      float re = s_re[base + k], im = s_im[base + k];
      float p = re * re + im * im;
      z0 += ((k >> 4) & 1) ? -p : p;
      z1 += ((k >> 3) & 1) ? -p : p;
      z2 += ((k >> 2) & 1) ? -p : p;
      z3 += ((k >> 1) & 1) ? -p : p;
      z4 += (k & 1)        ? -p : p;
    }
  }
  out[s] = readout_b[0] + readout_w[0] * z0 + readout_w[1] * z1 +
           readout_w[2] * z2 + readout_w[3] * z3 + readout_w[4] * z4;
}

// ---------------------------------------------------------------------------
extern "C" void kernel_launch(void* const* d_in, const int* in_sizes, int n_in,
                              void* d_out, int out_size, void* d_ws, size_t ws_size,
                              hipStream_t stream) {
  const float* x_short   = (const float*)d_in[0];
  const float* x_long    = (const float*)d_in[1];
  const float* conv_s_w  = (const float*)d_in[2];
  const float* conv_s_b  = (const float*)d_in[3];
  const float* lin_s_w   = (const float*)d_in[4];
  const float* lin_s_b   = (const float*)d_in[5];
  const float* conv_l_w  = (const float*)d_in[6];
  const float* conv_l_b  = (const float*)d_in[7];
  const float* lin_l_w   = (const float*)d_in[8];
  const float* lin_l_b   = (const float*)d_in[9];
  const float* res_w     = (const float*)d_in[10];
  const float* readout_w = (const float*)d_in[11];
  const float* readout_b = (const float*)d_in[12];
  float* out = (float*)d_out;

  float* Bre = (float*)d_ws;          // 3 * 32 * 32 floats
  float* Bim = Bre + 3 * 1024;        // 3 * 32 * 32 floats (24 KB total)

  build_gates<<<12, 256, 0, stream>>>(res_w, Bre, Bim);

  int B = in_sizes[0] / 5;            // samples (524288)
  msqcr_kernel<<<B / 128, 128, 0, stream>>>(
      x_short, x_long, conv_s_w, conv_s_b, lin_s_w, lin_s_b,
      conv_l_w, conv_l_b, lin_l_w, lin_l_b, readout_w, readout_b,
      Bre, Bim, out);
}